// Test_66881230733940
// MI455X (gfx1250) — compile-verified
//
#include <hip/hip_runtime.h>

// out[mask_idx[row]] += x[row]*y[row] + z[row], rows of D=64 f32.
// Roofline: 2*N*D = 128 MFLOP vs ~1 GB moved -> 0.12 FLOP/B: pure bandwidth.
// Strategy: NT b128 streaming loads for x/y/z (768 MB, touched once),
// RT zero-init so the 128 MB output stays dirty-resident in the 192 MB L2,
// scatter via native no-return global_atomic_add_f32 (L2-hit RMWs).

typedef float v4f __attribute__((ext_vector_type(4)));

#define TPB 256

__global__ __launch_bounds__(TPB) void fma_scatter_kernel(
    const v4f* __restrict__ x, const v4f* __restrict__ y,
    const v4f* __restrict__ z, const int* __restrict__ idx,
    float* __restrict__ out, int nquads) {
    int t = (int)(blockIdx.x * TPB + threadIdx.x);
    if (t >= nquads) return;

    // D = 64 floats per row -> 16 float4 quads per row; 16 lanes cover a row.
    unsigned row = (unsigned)t >> 4;
    unsigned q   = (unsigned)t & 15u;

    // Non-temporal 128-bit streaming loads (saddr + 32-bit voffset addressing).
    v4f a = __builtin_nontemporal_load(&x[(unsigned)t]);
    v4f b = __builtin_nontemporal_load(&y[(unsigned)t]);
    v4f c = __builtin_nontemporal_load(&z[(unsigned)t]);
    int orow = __builtin_nontemporal_load(&idx[row]);

    v4f v;
    v.x = __builtin_fmaf(a.x, b.x, c.x);
    v.y = __builtin_fmaf(a.y, b.y, c.y);
    v.z = __builtin_fmaf(a.z, b.z, c.z);
    v.w = __builtin_fmaf(a.w, b.w, c.w);

    // Byte offset fits in u32 (output is 128 MB): 32-bit offset addressing.
    unsigned off = ((unsigned)orow << 6) + (q << 2);
    float* p = out + off;
    // Relaxed agent-scope FP atomics -> global_atomic_add_f32, no return.
    __hip_atomic_fetch_add(&p[0], v.x, __ATOMIC_RELAXED, __HIP_MEMORY_SCOPE_AGENT);
    __hip_atomic_fetch_add(&p[1], v.y, __ATOMIC_RELAXED, __HIP_MEMORY_SCOPE_AGENT);
    __hip_atomic_fetch_add(&p[2], v.z, __ATOMIC_RELAXED, __HIP_MEMORY_SCOPE_AGENT);
    __hip_atomic_fetch_add(&p[3], v.w, __ATOMIC_RELAXED, __HIP_MEMORY_SCOPE_AGENT);
}

__global__ __launch_bounds__(TPB) void zero_out_kernel(v4f* __restrict__ out,
                                                       int n4) {
    int i = (int)(blockIdx.x * TPB + threadIdx.x);
    if (i < n4) {
        v4f zero = {0.0f, 0.0f, 0.0f, 0.0f};
        out[(unsigned)i] = zero;   // RT store: leaves output resident in L2
    }
}

extern "C" void kernel_launch(void* const* d_in, const int* in_sizes, int n_in,
                              void* d_out, int out_size, void* d_ws, size_t ws_size,
                              hipStream_t stream) {
    const v4f* x   = (const v4f*)d_in[0];
    const v4f* y   = (const v4f*)d_in[1];
    const v4f* z   = (const v4f*)d_in[2];
    const int* idx = (const int*)d_in[3];   // harness passes integers as int32
    float*     out = (float*)d_out;

    int nelem  = in_sizes[0];      // N*D = 64M, fits int32
    int nquads = nelem >> 2;       // float4 quads over x/y/z
    int out4   = out_size >> 2;    // float4 quads over output

    unsigned zb = (unsigned)((out4 + TPB - 1) / TPB);
    zero_out_kernel<<<zb, TPB, 0, stream>>>((v4f*)out, out4);

    unsigned sb = (unsigned)((nquads + TPB - 1) / TPB);
    fma_scatter_kernel<<<sb, TPB, 0, stream>>>(x, y, z, idx, out, nquads);
}